// HeteroDotProductPredictor_9466107920698
// MI455X (gfx1250) — compile-verified
//
#include <hip/hip_runtime.h>

typedef __attribute__((ext_vector_type(2))) float v2f;
typedef __attribute__((ext_vector_type(4))) float v4f;
typedef __attribute__((ext_vector_type(8))) float v8f;

#define D_FEAT   128
#define ROW_PAD  132   // 128 + 4 floats of padding: fragment reads hit 64 distinct banks
#define WAVES_PB 8
#define EPG      16    // edges per group; one wave32 per group

__global__ __launch_bounds__(WAVES_PB * 32)
void edge_dot_wmma_f32(const float* __restrict__ h,
                       const long long* __restrict__ src,
                       const long long* __restrict__ dst,
                       float* __restrict__ out,
                       int n_edges)
{
    __shared__ float tile[WAVES_PB * 2 * EPG * ROW_PAD];   // 132 KB

    const int lane  = threadIdx.x & 31;
    const int wave  = threadIdx.x >> 5;
    const int group = blockIdx.x * WAVES_PB + wave;
    const int ebase = group * EPG;
    if (ebase >= n_edges) return;            // wave-uniform exit: EXEC stays all-ones

    float* aRows = &tile[wave * (2 * EPG * ROW_PAD)];
    float* bRows = aRows + EPG * ROW_PAD;

    // ---- Stage 16 src rows + 16 dst rows into LDS via the CDNA5 async path ----
    // global_load_async_to_lds_b128: per-lane 16B memory->LDS, no VGPR bounce,
    // tracked by ASYNCcnt. One instruction moves a full coalesced 512B row.
    const int c4 = lane * 4;                 // this lane's 4-float slice of the row
    for (int r = 0; r < EPG; ++r) {
        int e = ebase + r;
        if (e >= n_edges) e = n_edges - 1;   // tail clamp keeps all lanes active
        const long long si = src[e];
        const long long di = dst[e];
        const float* gs = h + si * (long long)D_FEAT + c4;
        const float* gd = h + di * (long long)D_FEAT + c4;
        // LDS aperture: low 32 bits of the generic pointer == wave-relative LDS byte offset
        unsigned ldsA = (unsigned)(unsigned long long)(aRows + r * ROW_PAD + c4);
        unsigned ldsB = (unsigned)(unsigned long long)(bRows + r * ROW_PAD + c4);
        asm volatile("global_load_async_to_lds_b128 %0, %1, off"
                     :: "v"(ldsA), "v"(gs) : "memory");
        asm volatile("global_load_async_to_lds_b128 %0, %1, off"
                     :: "v"(ldsB), "v"(gd) : "memory");
    }
    asm volatile("s_wait_asynccnt 0x0" ::: "memory");
    // Wave-private LDS region => no barrier; fragment ds_loads below cannot be
    // hoisted above the wait (memory clobbers).

    // ---- Fragment addressing (identical lane pattern for A 16x4 and B 4x16 f32) ----
    // A: lanes 0-15 hold M=lane, K = k0+{0,1}; lanes 16-31 hold M=lane-16, K = k0+{2,3}.
    // B: lanes 0-15 hold N=lane, K = k0+{0,1}; lanes 16-31 hold N=lane-16, K = k0+{2,3}.
    const int m    = lane & 15;
    const int koff = (lane >> 4) * 2;
    const float* aF = aRows + m * ROW_PAD + koff;
    const float* bF = bRows + m * ROW_PAD + koff;

    v8f acc = {0.f, 0.f, 0.f, 0.f, 0.f, 0.f, 0.f, 0.f};
#pragma unroll
    for (int k0 = 0; k0 < D_FEAT; k0 += 4) {
        v2f a = *(const v2f*)(aF + k0);       // ds_load_2addr_b64, bank-conflict-free
        v2f b = *(const v2f*)(bF + k0);
        // D = A x B + C ; diagonal D[e][e] accumulates the per-edge dot product.
        acc = __builtin_amdgcn_wmma_f32_16x16x4_f32(false, a, false, b,
                                                    (short)0, acc, false, false);
    }

    // ---- Diagonal extraction ----
    // C/D layout: lane L, VGPR r -> M = r + 8*(L>=16), N = L%16.
    // Diag lives in: lanes 0-7 (edge=lane, r=lane) and lanes 24-31 (edge=lane-16, r=lane-24).
    int sel = -1, eidx = 0;
    if (lane < 8)        { sel = lane;      eidx = lane;      }
    else if (lane >= 24) { sel = lane - 24; eidx = lane - 16; }
    if (sel >= 0) {
        float v = acc[0];
#pragma unroll
        for (int i = 1; i < 8; ++i) if (sel == i) v = acc[i];
        const int e = ebase + eidx;
        if (e < n_edges) out[e] = v;
    }
}

extern "C" void kernel_launch(void* const* d_in, const int* in_sizes, int n_in,
                              void* d_out, int out_size, void* d_ws, size_t ws_size,
                              hipStream_t stream) {
    const float* h       = (const float*)d_in[0];
    const long long* src = (const long long*)d_in[1];
    const long long* dst = (const long long*)d_in[2];
    float* out           = (float*)d_out;
    const int n_edges    = in_sizes[1];

    const int groups = (n_edges + EPG - 1) / EPG;
    const int blocks = (groups + WAVES_PB - 1) / WAVES_PB;
    edge_dot_wmma_f32<<<blocks, WAVES_PB * 32, 0, stream>>>(h, src, dst, out, n_edges);
}